// VecPointNet_70334384439615
// MI455X (gfx1250) — compile-verified
//
#include <hip/hip_runtime.h>
#include <hip/hip_bf16.h>

#define NPT   4096          // N points
#define RS    12288         // 3*N, channel stride in (B, C, 3, N) layout
#define NK    16            // K nearest neighbors
#define HCH   128           // hidden channels
#define LNA_EPS 1e-6f

typedef float v2f __attribute__((ext_vector_type(2)));
typedef float v8f __attribute__((ext_vector_type(8)));

// ---------------------------------------------------------------------------
// Kernel 1: KNN — one thread per (b, n); register-resident insertion sort.
// ---------------------------------------------------------------------------
__global__ __launch_bounds__(256) void knn_kernel(const float* __restrict__ x,
                                                  int* __restrict__ knn) {
  int t = blockIdx.x * 256 + threadIdx.x;       // t < 2*4096
  int b = t >> 12;
  int n = t & (NPT - 1);
  const float* xb = x + (size_t)b * 3 * NPT;
  float px = xb[n], py = xb[NPT + n], pz = xb[2 * NPT + n];

  float bd[NK];
  int   bi[NK];
#pragma unroll
  for (int i = 0; i < NK; ++i) { bd[i] = 3.4e38f; bi[i] = 0; }

  for (int m = 0; m < NPT; ++m) {
    float dx = px - xb[m];
    float dy = py - xb[NPT + m];
    float dz = pz - xb[2 * NPT + m];
    float d2 = dx * dx + dy * dy + dz * dz;
    if (d2 < bd[NK - 1]) {
      // fully-unrolled stable insertion (keeps arrays in VGPRs)
#pragma unroll
      for (int i = NK - 1; i >= 1; --i) {
        bool shift = d2 < bd[i - 1];
        bool here  = !shift && (d2 < bd[i]);
        bd[i] = shift ? bd[i - 1] : (here ? d2 : bd[i]);
        bi[i] = shift ? bi[i - 1] : (here ? m  : bi[i]);
      }
      if (d2 < bd[0]) { bd[0] = d2; bi[0] = m; }
    }
  }
#pragma unroll
  for (int k = 0; k < NK; ++k) knn[(size_t)t * NK + k] = bi[k];
}

// ---------------------------------------------------------------------------
// Kernel 2: input layer — build [cross, nb-x, x] features, apply W_in/U_in LNA,
// mean over K.  One block (128 threads = one per output channel) per (b, n).
// ---------------------------------------------------------------------------
__global__ __launch_bounds__(128) void input_layer_kernel(
    const float* __restrict__ x, const int* __restrict__ knn,
    const float* __restrict__ W_in, const float* __restrict__ U_in,
    float* __restrict__ hbuf) {
  int bn = blockIdx.x;                   // b*4096 + n
  int b = bn >> 12;
  int n = bn & (NPT - 1);
  const float* xb = x + (size_t)b * 3 * NPT;
  float px = xb[n], py = xb[NPT + n], pz = xb[2 * NPT + n];

  __shared__ float sc[NK][3];   // cross(x_dir, nb)
  __shared__ float sd[NK][3];   // nb - x
  if (threadIdx.x < NK) {
    int k = threadIdx.x;
    int j = knn[(size_t)bn * NK + k];
    float nx = xb[j], ny = xb[NPT + j], nz = xb[2 * NPT + j];
    float pn  = sqrtf(px * px + py * py + pz * pz);
    float inv = 1.0f / fmaxf(pn, 1e-12f);
    float ax = px * inv, ay = py * inv, az = pz * inv;
    sc[k][0] = ay * nz - az * ny;
    sc[k][1] = az * nx - ax * nz;
    sc[k][2] = ax * ny - ay * nx;
    sd[k][0] = nx - px; sd[k][1] = ny - py; sd[k][2] = nz - pz;
  }
  __syncthreads();

  int o = threadIdx.x;
  float w0 = W_in[o * 3], w1 = W_in[o * 3 + 1], w2 = W_in[o * 3 + 2];
  float u0 = U_in[o * 3], u1 = U_in[o * 3 + 1], u2 = U_in[o * 3 + 2];
  float a0 = 0.f, a1 = 0.f, a2 = 0.f;
#pragma unroll
  for (int k = 0; k < NK; ++k) {
    float P0 = w0 * sc[k][0] + w1 * sd[k][0] + w2 * px;
    float P1 = w0 * sc[k][1] + w1 * sd[k][1] + w2 * py;
    float P2 = w0 * sc[k][2] + w1 * sd[k][2] + w2 * pz;
    float D0 = u0 * sc[k][0] + u1 * sd[k][0] + u2 * px;
    float D1 = u0 * sc[k][1] + u1 * sd[k][1] + u2 * py;
    float D2 = u0 * sc[k][2] + u1 * sd[k][2] + u2 * pz;
    float dot = P0 * D0 + P1 * D1 + P2 * D2;
    float dsq = D0 * D0 + D1 * D1 + D2 * D2;
    float s = (dot >= 0.f) ? 0.f : dot / (dsq + LNA_EPS);
    a0 += P0 - s * D0; a1 += P1 - s * D1; a2 += P2 - s * D2;
  }
  size_t base = ((size_t)b * HCH + o) * RS + n;
  const float ik = 1.0f / NK;
  hbuf[base]            = a0 * ik;
  hbuf[base + NPT]      = a1 * ik;
  hbuf[base + 2 * NPT]  = a2 * ik;
}

// ---------------------------------------------------------------------------
// Kernel 3: dual GEMM  P = W@X (+biasP),  D = U@X (+biasU)  via f32 WMMA.
// Block = 128 thr (4 waves) arranged 2x2; each wave owns a 32x32 output
// macro-tile (2x2 of 16x16 WMMA tiles), so per K-step it issues 4 A-fragment
// loads + 2 B-fragment loads feeding 8 v_wmma_f32_16x16x4_f32 (HAS_U case).
// Block covers 64 out-channels x 64 columns.  Columns are (b,d,n)-ordered;
// input channel stride is RS.
// ---------------------------------------------------------------------------
template <bool HAS_U>
__global__ __launch_bounds__(128) void gemm_pd_wmma(
    const float* __restrict__ W, const float* __restrict__ U, int ldw,
    const float* __restrict__ X, size_t ldx_batch,
    const float* __restrict__ biasP, const float* __restrict__ biasU,
    float* __restrict__ P, float* __restrict__ D, int Kdim) {
  const int wave = threadIdx.x >> 5;
  const int lane = threadIdx.x & 31;
  const int hf   = lane >> 4;               // which half of the wave
  const int lr   = lane & 15;
  const int wr   = wave >> 1;               // wave row (channel group)
  const int wc   = wave & 1;                // wave col (column group)
  const int chBase  = blockIdx.x * 64 + wr * 32;   // + {0,16}
  const int colBase = blockIdx.y * 64 + wc * 32;   // + {0,16}
  const int b  = colBase / RS;
  const int cc = colBase - b * RS;          // column within batch [0, 3N)
  const int dIdx = cc >> 12;                // xyz component (uniform: 4096%64==0)

  const float* Wrow0 = W + (size_t)(chBase + lr) * ldw;
  const float* Wrow1 = W + (size_t)(chBase + 16 + lr) * ldw;
  const float* Urow0 = HAS_U ? (U + (size_t)(chBase + lr) * ldw) : nullptr;
  const float* Urow1 = HAS_U ? (U + (size_t)(chBase + 16 + lr) * ldw) : nullptr;
  const float* Xcol0 = X + (size_t)b * ldx_batch + cc + lr;
  const float* Xcol1 = Xcol0 + 16;

  const v8f vz = {0.f, 0.f, 0.f, 0.f, 0.f, 0.f, 0.f, 0.f};
  v8f accP00 = vz, accP01 = vz, accP10 = vz, accP11 = vz;
  v8f accD00 = vz, accD01 = vz, accD10 = vz, accD11 = vz;

#pragma unroll 2
  for (int k0 = 0; k0 < Kdim; k0 += 4) {
    const int kk = k0 + hf * 2;
    // A fragments: lanes 0-15 rows M, K={k0,k0+1}; lanes 16-31 K={k0+2,k0+3}
    v2f aW0 = *(const v2f*)(Wrow0 + kk);
    v2f aW1 = *(const v2f*)(Wrow1 + kk);
    // B fragments mirror A's K split; column = lane&15, row stride = RS
    v2f b0, b1;
    b0.x = Xcol0[(size_t)kk * RS];
    b0.y = Xcol0[(size_t)(kk + 1) * RS];
    b1.x = Xcol1[(size_t)kk * RS];
    b1.y = Xcol1[(size_t)(kk + 1) * RS];
    accP00 = __builtin_amdgcn_wmma_f32_16x16x4_f32(false, aW0, false, b0,
                                                   (short)0, accP00, false, false);
    accP01 = __builtin_amdgcn_wmma_f32_16x16x4_f32(false, aW0, false, b1,
                                                   (short)0, accP01, false, false);
    accP10 = __builtin_amdgcn_wmma_f32_16x16x4_f32(false, aW1, false, b0,
                                                   (short)0, accP10, false, false);
    accP11 = __builtin_amdgcn_wmma_f32_16x16x4_f32(false, aW1, false, b1,
                                                   (short)0, accP11, false, false);
    if (HAS_U) {
      v2f aU0 = *(const v2f*)(Urow0 + kk);
      v2f aU1 = *(const v2f*)(Urow1 + kk);
      accD00 = __builtin_amdgcn_wmma_f32_16x16x4_f32(false, aU0, false, b0,
                                                     (short)0, accD00, false, false);
      accD01 = __builtin_amdgcn_wmma_f32_16x16x4_f32(false, aU0, false, b1,
                                                     (short)0, accD01, false, false);
      accD10 = __builtin_amdgcn_wmma_f32_16x16x4_f32(false, aU1, false, b0,
                                                     (short)0, accD10, false, false);
      accD11 = __builtin_amdgcn_wmma_f32_16x16x4_f32(false, aU1, false, b1,
                                                     (short)0, accD11, false, false);
    }
  }

  auto store_tile = [&](const v8f& acc, const float* __restrict__ bias,
                        float* __restrict__ out, int ch0, int c0) {
    size_t colAddr = (size_t)b * ((size_t)HCH * RS) + c0 + lr;
#pragma unroll
    for (int r = 0; r < 8; ++r) {
      int o = ch0 + hf * 8 + r;             // C/D layout: VGPR r -> M = r + 8*hf
      float v = acc[r];
      if (bias) v += bias[(b * 3 + dIdx) * HCH + o];
      out[colAddr + (size_t)o * RS] = v;
    }
  };
  store_tile(accP00, biasP, P, chBase,      cc);
  store_tile(accP01, biasP, P, chBase,      cc + 16);
  store_tile(accP10, biasP, P, chBase + 16, cc);
  store_tile(accP11, biasP, P, chBase + 16, cc + 16);
  if (HAS_U) {
    store_tile(accD00, biasU, D, chBase,      cc);
    store_tile(accD01, biasU, D, chBase,      cc + 16);
    store_tile(accD10, biasU, D, chBase + 16, cc);
    store_tile(accD11, biasU, D, chBase + 16, cc + 16);
  }
}

// ---------------------------------------------------------------------------
// Kernel 4: elementwise LNA — couples the three xyz columns per (b, o, n).
// ---------------------------------------------------------------------------
__global__ __launch_bounds__(256) void lna_kernel(const float* __restrict__ P,
                                                  const float* __restrict__ Dm,
                                                  float* __restrict__ Y,
                                                  size_t y_batch_stride) {
  int t = blockIdx.x * 256 + threadIdx.x;   // t < 2*128*4096
  int n = t & (NPT - 1);
  int o = (t >> 12) & (HCH - 1);
  int b = t >> 19;
  size_t base = ((size_t)b * HCH + o) * RS + n;
  float p0 = P[base], p1 = P[base + NPT], p2 = P[base + 2 * NPT];
  float d0 = Dm[base], d1 = Dm[base + NPT], d2 = Dm[base + 2 * NPT];
  float dot = p0 * d0 + p1 * d1 + p2 * d2;
  float dsq = d0 * d0 + d1 * d1 + d2 * d2;
  float s = (dot >= 0.f) ? 0.f : dot / (dsq + LNA_EPS);
  size_t yb = (size_t)b * y_batch_stride + (size_t)o * RS + n;
  Y[yb]           = p0 - s * d0;
  Y[yb + NPT]     = p1 - s * d1;
  Y[yb + 2 * NPT] = p2 - s * d2;
}

// ---------------------------------------------------------------------------
// Kernel 5: row mean over N=4096 — one block per (b, c, d) row; used both for
// yg (layer global feature) and the final output mean.
// ---------------------------------------------------------------------------
__global__ __launch_bounds__(256) void row_mean_kernel(const float* __restrict__ src,
                                                       float* __restrict__ dst) {
  __shared__ float red[256];
  int r = blockIdx.x;                       // r = (b*128 + c)*3 + d
  const float* p = src + (size_t)r * NPT;
  float s = 0.f;
  for (int i = threadIdx.x; i < NPT; i += 256) s += p[i];
  red[threadIdx.x] = s;
  __syncthreads();
  for (int off = 128; off > 0; off >>= 1) {
    if (threadIdx.x < off) red[threadIdx.x] += red[threadIdx.x + off];
    __syncthreads();
  }
  if (threadIdx.x == 0) dst[r] = red[0] * (1.0f / NPT);
}

// ---------------------------------------------------------------------------
// Kernel 6: bias from broadcast yg half of Wg/Ug:
//   biasP[(b*3+d)*128 + o] = sum_c Wg[o, 128+c] * yg[b, c, d]
// ---------------------------------------------------------------------------
__global__ __launch_bounds__(256) void gbias_kernel(const float* __restrict__ Wg,
                                                    const float* __restrict__ Ug,
                                                    const float* __restrict__ yg,
                                                    float* __restrict__ biasP,
                                                    float* __restrict__ biasU) {
  int t = blockIdx.x * 256 + threadIdx.x;   // t < 768
  int o  = t & (HCH - 1);
  int bd = t >> 7;                          // b*3 + d
  int b = bd / 3, d = bd - b * 3;
  const float* wrow = Wg + (size_t)o * 256 + 128;
  const float* urow = Ug + (size_t)o * 256 + 128;
  float sp = 0.f, su = 0.f;
  for (int c = 0; c < HCH; ++c) {
    float yv = yg[(b * HCH + c) * 3 + d];
    sp += wrow[c] * yv;
    su += urow[c] * yv;
  }
  biasP[bd * HCH + o] = sp;
  biasU[bd * HCH + o] = su;
}

// ---------------------------------------------------------------------------
extern "C" void kernel_launch(void* const* d_in, const int* in_sizes, int n_in,
                              void* d_out, int out_size, void* d_ws, size_t ws_size,
                              hipStream_t stream) {
  (void)in_sizes; (void)n_in; (void)out_size; (void)ws_size;

  const float* x     = (const float*)d_in[0];   // (2, 3, 4096)
  const float* W_in  = (const float*)d_in[1];   // (128, 3)
  const float* U_in  = (const float*)d_in[2];
  const float* Wl    = (const float*)d_in[3];   // (4, 128, 128)
  const float* Ul    = (const float*)d_in[4];
  const float* Wg    = (const float*)d_in[5];   // (4, 128, 256)
  const float* Ug    = (const float*)d_in[6];
  const float* W_out = (const float*)d_in[7];   // (128, 512)
  float* out = (float*)d_out;                   // [768 mean][2*128*3*4096 out]

  // workspace layout
  int*   knn  = (int*)d_ws;                             // 2*4096*16 ints
  float* fb   = (float*)d_ws + (size_t)2 * NPT * NK;
  const size_t SZ = (size_t)2 * HCH * RS;               // (B,128,3,N) floats
  float* hbuf  = fb;                                    // mid-layer y
  float* Pb    = hbuf + SZ;
  float* Db    = Pb + SZ;
  float* feats = Db + SZ;                               // (B,512,3,N) concat
  float* yg    = feats + (size_t)2 * 512 * RS;          // 768
  float* biasP = yg + 768;                              // 768
  float* biasU = biasP + 768;                           // 768

  knn_kernel<<<32, 256, 0, stream>>>(x, knn);
  input_layer_kernel<<<2 * NPT, 128, 0, stream>>>(x, knn, W_in, U_in, hbuf);

  const dim3 ggrid(2, (2 * RS) / 64);   // 2 ch-groups x 384 column macro-tiles
  for (int i = 0; i < 4; ++i) {
    const float* Wi  = Wl + (size_t)i * HCH * HCH;
    const float* Ui  = Ul + (size_t)i * HCH * HCH;
    const float* Wgi = Wg + (size_t)i * HCH * 256;
    const float* Ugi = Ug + (size_t)i * HCH * 256;
    const float* Xin = (i == 0) ? hbuf : (feats + (size_t)(i - 1) * HCH * RS);
    size_t ldxb = (i == 0) ? (size_t)HCH * RS : (size_t)512 * RS;

    gemm_pd_wmma<true><<<ggrid, 128, 0, stream>>>(Wi, Ui, HCH, Xin, ldxb,
                                                  nullptr, nullptr, Pb, Db, HCH);
    lna_kernel<<<4096, 256, 0, stream>>>(Pb, Db, hbuf, (size_t)HCH * RS);
    row_mean_kernel<<<768, 256, 0, stream>>>(hbuf, yg);
    gbias_kernel<<<3, 256, 0, stream>>>(Wgi, Ugi, yg, biasP, biasU);
    gemm_pd_wmma<true><<<ggrid, 128, 0, stream>>>(Wgi, Ugi, 256, hbuf,
                                                  (size_t)HCH * RS,
                                                  biasP, biasU, Pb, Db, HCH);
    lna_kernel<<<4096, 256, 0, stream>>>(Pb, Db, feats + (size_t)i * HCH * RS,
                                         (size_t)512 * RS);
  }

  // final linear: out = W_out @ feats   (128 x 512) x (512 x 24576)
  gemm_pd_wmma<false><<<ggrid, 128, 0, stream>>>(W_out, nullptr, 512, feats,
                                                 (size_t)512 * RS, nullptr,
                                                 nullptr, out + 768, nullptr, 512);
  // mean over N into the first 768 outputs
  row_mean_kernel<<<768, 256, 0, stream>>>(out + 768, out);
}